// splineconv_3667902071139
// MI455X (gfx1250) — compile-verified
//
#include <hip/hip_runtime.h>
#include <hip/hip_bf16.h>
#include <math.h>

typedef __attribute__((ext_vector_type(2))) float v2f;
typedef __attribute__((ext_vector_type(8))) float v8f;

// ---------------------------------------------------------------------------
// Kernel 1: Y{0,1,2}[N x FOUT] = X[N x 32] @ B{0,1,2}[32 x FOUT]
// One wave per 16-row tile. f32 WMMA 16x16x4, K accumulated over 8 steps.
// A layout (16x4 f32): lane = M (l&15), VGPR j holds K = j + 2*(lane>>4).
// B layout (4x16 f32): lane = N (l&15), VGPR j holds K = j + 2*(lane>>4).
// C/D layout (16x16 f32): VGPR j, lanes 0-15 -> M=j, lanes 16-31 -> M=j+8.
// ---------------------------------------------------------------------------
template <int FOUT>
__global__ void gemm3_wmma_kernel(const float* __restrict__ X,
                                  const float* __restrict__ B0,
                                  const float* __restrict__ B1,
                                  const float* __restrict__ B2,
                                  float* __restrict__ Y0,
                                  float* __restrict__ Y1,
                                  float* __restrict__ Y2,
                                  int nTiles)
{
    int wave = blockIdx.x * (blockDim.x >> 5) + (threadIdx.x >> 5);
    if (wave >= nTiles) return;             // wave-uniform: EXEC stays all-1s
    const int lane = threadIdx.x & 31;
    const int half = lane >> 4;             // 0 or 1
    const int l16  = lane & 15;
    const int m0   = wave * 16;

    // Load the 16x32 A tile once: 8 K-chunks of a contiguous float pair/lane.
    v2f a[8];
    const float* xrow = X + (size_t)(m0 + l16) * 32 + 2 * half;
    #pragma unroll
    for (int k0 = 0; k0 < 8; ++k0)
        a[k0] = *(const v2f*)(xrow + k0 * 4);

    const float* Bm[3] = {B0, B1, B2};
    float*       Ym[3] = {Y0, Y1, Y2};

    #pragma unroll
    for (int rsel = 0; rsel < 3; ++rsel) {
        const float* W = Bm[rsel];
        float*       Y = Ym[rsel];
        #pragma unroll
        for (int ct = 0; ct < FOUT / 16; ++ct) {
            const int col = ct * 16 + l16;
            v8f c = {};
            #pragma unroll
            for (int k0 = 0; k0 < 8; ++k0) {
                const int kb = k0 * 4 + 2 * half;
                v2f b;
                b.x = W[(kb    ) * FOUT + col];
                b.y = W[(kb + 1) * FOUT + col];
                // 8 args: (neg_a, A, neg_b, B, c_mod, C, reuse_a, reuse_b)
                c = __builtin_amdgcn_wmma_f32_16x16x4_f32(
                        false, a[k0], false, b, (short)0, c, false, false);
            }
            #pragma unroll
            for (int j = 0; j < 8; ++j)
                Y[(size_t)(m0 + j + 8 * half) * FOUT + col] = c[j];
        }
    }
}

// ---------------------------------------------------------------------------
// Kernel 2: edge gather + axpy + atomic scatter.
// Each wave owns 32 edges: lane L loads edge (e0+L)'s metadata once
// (coalesced, non-temporal), then the wave sweeps the 32 edges broadcasting
// (src,dst,u) via __shfl; every lane handles one output feature.
// F = 1<<logF features -> 32/F edges processed concurrently per sweep step.
// ---------------------------------------------------------------------------
template <int LOGF>
__global__ void edge_scatter_kernel(const long long* __restrict__ ei,
                                    const float* __restrict__ ea,
                                    const float* __restrict__ y0,
                                    const float* __restrict__ y1,
                                    float* __restrict__ agg,
                                    float* __restrict__ cnt,
                                    int E, int doCnt)
{
    constexpr int F   = 1 << LOGF;
    constexpr int EPI = 32 >> LOGF;     // edges per sweep step (1 or 2)

    const int lane = threadIdx.x & 31;
    const long long wave =
        (long long)blockIdx.x * (blockDim.x >> 5) + (threadIdx.x >> 5);
    const long long e0 = wave * 32;

    // One edge's metadata per lane (read-once streams: non-temporal).
    int   src = 0, dst = 0;
    float u   = 0.0f;
    {
        const long long e = e0 + lane;
        if (e < E) {
            src = (int)__builtin_nontemporal_load(&ei[e]);
            dst = (int)__builtin_nontemporal_load(&ei[(size_t)E + e]);
            float uu = __builtin_nontemporal_load(&ea[e]);
            u = fminf(fmaxf(uu, 0.0f), 1.0f);
        }
    }

    const int sub = lane >> LOGF;       // which edge of the EPI-group
    const int f   = lane & (F - 1);     // feature handled by this lane

    #pragma unroll 4
    for (int i = 0; i < 32; i += EPI) {
        const int   idx = i + sub;      // source lane for metadata
        const int   s   = __shfl(src, idx);
        const int   d   = __shfl(dst, idx);
        const float uu  = __shfl(u,   idx);
        if (e0 + idx < E) {
            const size_t si = (size_t)s * F + f;
            const float  v  = (1.0f - uu) * y0[si] + uu * y1[si];
            atomicAdd(&agg[(size_t)d * F + f], v);
            if (doCnt && f == 0) atomicAdd(&cnt[d], 1.0f);
        }
    }
}

// ---------------------------------------------------------------------------
// Kernel 3: layer-1 finalize  h = relu(agg/max(cnt,1) + r + b)   (F = 32)
// ---------------------------------------------------------------------------
__global__ void finalize1_kernel(const float* __restrict__ agg,
                                 const float* __restrict__ cnt,
                                 const float* __restrict__ r,
                                 const float* __restrict__ b,
                                 float* __restrict__ h, int N)
{
    int t = blockIdx.x * blockDim.x + threadIdx.x;
    if (t >= N * 32) return;
    const int i = t >> 5, f = t & 31;
    const float c = fmaxf(cnt[i], 1.0f);
    const float v = agg[t] / c + r[t] + b[f];
    h[t] = fmaxf(v, 0.0f);
}

// ---------------------------------------------------------------------------
// Kernel 4: layer-2 finalize + log_softmax over 16 features (register file)
// ---------------------------------------------------------------------------
__global__ void finalize2_kernel(const float* __restrict__ agg,
                                 const float* __restrict__ cnt,
                                 const float* __restrict__ r,
                                 const float* __restrict__ b,
                                 float* __restrict__ out, int N)
{
    int i = blockIdx.x * blockDim.x + threadIdx.x;
    if (i >= N) return;
    const float inv = 1.0f / fmaxf(cnt[i], 1.0f);
    float z[16];
    float m = -INFINITY;
    #pragma unroll
    for (int f = 0; f < 16; ++f) {
        z[f] = agg[(size_t)i * 16 + f] * inv + r[(size_t)i * 16 + f] + b[f];
        m = fmaxf(m, z[f]);
    }
    float s = 0.0f;
    #pragma unroll
    for (int f = 0; f < 16; ++f) s += __expf(z[f] - m);
    const float ls = __logf(s) + m;
    #pragma unroll
    for (int f = 0; f < 16; ++f) out[(size_t)i * 16 + f] = z[f] - ls;
}

// ---------------------------------------------------------------------------
extern "C" void kernel_launch(void* const* d_in, const int* in_sizes, int n_in,
                              void* d_out, int out_size, void* d_ws, size_t ws_size,
                              hipStream_t stream)
{
    const float*     x     = (const float*)d_in[0];
    const long long* ei    = (const long long*)d_in[1];
    const float*     ea    = (const float*)d_in[2];
    const float*     W1    = (const float*)d_in[3];   // [2,32,32]
    const float*     root1 = (const float*)d_in[4];   // [32,32]
    const float*     b1    = (const float*)d_in[5];   // [32]
    const float*     W2    = (const float*)d_in[6];   // [2,32,16]
    const float*     root2 = (const float*)d_in[7];   // [32,16]
    const float*     b2    = (const float*)d_in[8];   // [16]
    float* out = (float*)d_out;

    const int N = in_sizes[0] / 32;     // 100000
    const int E = in_sizes[2];          // 3200000

    // Workspace layout (floats)
    float* ws = (float*)d_ws;
    const size_t NF32 = (size_t)N * 32;
    float* y0  = ws;                    // [N,32] (layer2 reuses [N,16] region)
    float* y1  = y0  + NF32;
    float* rr  = y1  + NF32;            // x@root
    float* agg = rr  + NF32;
    float* cnt = agg + NF32;            // [N]
    float* h   = cnt + N;               // [N,32]

    const int nTiles = N / 16;          // 6250, exact (100000 % 16 == 0)
    const int wavesPerBlock = 4;        // 128 threads = 4 wave32
    const int gemmBlocks = (nTiles + wavesPerBlock - 1) / wavesPerBlock;

    const long long edgeWaves  = ((long long)E + 31) / 32;
    const int       edgeBlocks = (int)((edgeWaves + 7) / 8);   // 8 waves/block

    // ---------------- Layer 1 ----------------
    hipMemsetAsync(agg, 0, (NF32 + (size_t)N) * sizeof(float), stream); // agg+cnt

    gemm3_wmma_kernel<32><<<gemmBlocks, wavesPerBlock * 32, 0, stream>>>(
        x, W1, W1 + 32 * 32, root1, y0, y1, rr, nTiles);

    edge_scatter_kernel<5><<<edgeBlocks, 256, 0, stream>>>(
        ei, ea, y0, y1, agg, cnt, E, 1);

    finalize1_kernel<<<(N * 32 + 255) / 256, 256, 0, stream>>>(
        agg, cnt, rr, b1, h, N);

    // ---------------- Layer 2 ----------------
    hipMemsetAsync(agg, 0, (size_t)N * 16 * sizeof(float), stream);

    gemm3_wmma_kernel<16><<<gemmBlocks, wavesPerBlock * 32, 0, stream>>>(
        h, W2, W2 + 32 * 16, root2, y0, y1, rr, nTiles);

    edge_scatter_kernel<4><<<edgeBlocks, 256, 0, stream>>>(
        ei, ea, y0, y1, agg, cnt, E, 0);     // reuse degree from layer 1

    finalize2_kernel<<<(N + 255) / 256, 256, 0, stream>>>(
        agg, cnt, rr, b2, out, N);
}